// TCN_11828339933946
// MI455X (gfx1250) — compile-verified
//
#include <hip/hip_runtime.h>

#define LOG2L  12
#define LEN    4096
#define NBATCH 16
#define MTOT   (NBATCH * LEN)   // 65536 time rows
#define NCLS   5

typedef float v2f __attribute__((ext_vector_type(2)));
typedef float v8f __attribute__((ext_vector_type(8)));

// ---------------------------------------------------------------------------
// Dilated causal conv as TAPS shifted GEMMs via V_WMMA_F32_16X16X4_F32.
// y[m, co] = bias[co] + sum_tap sum_ci W[co, ci, tap] * x[b, t-(TAPS-1-tap)*dil, ci]
// One wave computes TWO 16(time) x 16(cout) tiles (B operand reused from LDS).
// CIN is compile time -> fully unrolled K loop, dense v_wmma stream.
// ---------------------------------------------------------------------------
template <int TAPS, int CIN, bool FULL>
__device__ __forceinline__ void conv_tile_compute(
    const float* __restrict__ x, const float* __restrict__ wlds,
    int mbase, int dil, int lane, v8f& acc0, v8f& acc1)
{
    const int khalf = lane >> 4;
    const int n     = lane & 15;
    const int b     = mbase >> LOG2L;
    const int t0    = (mbase & (LEN - 1)) + (lane & 15);   // tile0 time (per lane)
#pragma unroll
    for (int tap = 0; tap < TAPS; ++tap) {
        const int  shift = (TAPS - 1 - tap) * dil;
        const int  tt0   = t0 - shift;
        const int  tt1   = t0 + 16 - shift;                // >= 0 always (shift <= 16)
        const bool v0    = FULL || (tt0 >= 0);             // causal left pad
        const float* xr0 = x + (size_t)((b << LOG2L) + (v0 ? tt0 : 0)) * CIN;
        const float* xr1 = x + (size_t)((b << LOG2L) + tt1) * CIN;
        const float* wt  = wlds + tap * CIN * 16;
#pragma unroll
        for (int k0 = 0; k0 < CIN; k0 += 4) {
            const int kA0 = k0 + (khalf << 1);             // A: VGPRv = K v+2*(lane>=16)
            v2f a0, a1, bm;
            if (k0 + 3 < CIN) {                            // compile-time per unrolled iter
                a0 = *(const v2f*)(xr0 + kA0);             // aligned 8B loads (kA0 even,
                a1 = *(const v2f*)(xr1 + kA0);             //  row strides even)
                bm.x = wt[kA0 * 16 + n];
                bm.y = wt[(kA0 + 1) * 16 + n];
            } else {                                       // ragged tail (CIN=34 only)
                const int kA1 = kA0 + 1;
                a0.x = (kA0 < CIN) ? xr0[kA0] : 0.0f;
                a0.y = (kA1 < CIN) ? xr0[kA1] : 0.0f;
                a1.x = (kA0 < CIN) ? xr1[kA0] : 0.0f;
                a1.y = (kA1 < CIN) ? xr1[kA1] : 0.0f;
                bm.x = (kA0 < CIN) ? wt[kA0 * 16 + n] : 0.0f;
                bm.y = (kA1 < CIN) ? wt[kA1 * 16 + n] : 0.0f;
            }
            if (!FULL) {                                   // zero A rows left of t=0
                a0.x = v0 ? a0.x : 0.0f;
                a0.y = v0 ? a0.y : 0.0f;
            }
            acc0 = __builtin_amdgcn_wmma_f32_16x16x4_f32(false, a0, false, bm,
                                                         (short)0, acc0, false, false);
            acc1 = __builtin_amdgcn_wmma_f32_16x16x4_f32(false, a1, false, bm,
                                                         (short)0, acc1, false, false);
        }
    }
}

template <int TAPS, int CIN>
__global__ __launch_bounds__(128) void conv_wmma_kernel(
    const float* __restrict__ x, const float* __restrict__ w,
    const float* __restrict__ bias, float* __restrict__ y,
    int Cout, int dil)
{
    __shared__ float wlds[TAPS * CIN * 16];                // <= 24KB of 320KB WGP LDS
    const int tid   = threadIdx.x;
    const int lane  = tid & 31;
    const int wv    = tid >> 5;
    const int nbase = blockIdx.y << 4;

    for (int i = tid; i < TAPS * CIN * 16; i += 128) {
        const int n   = i & 15;
        const int ci  = (i >> 4) % CIN;
        const int tap = (i >> 4) / CIN;
        wlds[i] = w[(size_t)(nbase + n) * (CIN * TAPS) + ci * TAPS + tap];
    }
    __syncthreads();

    const int mbase = ((blockIdx.x << 2) + wv) << 5;       // 32 rows per wave
    v8f acc0 = {}, acc1 = {};
    const int tbase = mbase & (LEN - 1);
    if (tbase >= (TAPS - 1) * dil) {                       // uniform per wave: EXEC stays full
        conv_tile_compute<TAPS, CIN, true >(x, wlds, mbase, dil, lane, acc0, acc1);
    } else {
        conv_tile_compute<TAPS, CIN, false>(x, wlds, mbase, dil, lane, acc0, acc1);
    }

    const int   khalf = lane >> 4;
    const int   n     = lane & 15;
    const int   co    = nbase + n;
    const float bv    = bias[co];
#pragma unroll
    for (int v = 0; v < 8; ++v) {
        const int m0 = mbase + v + (khalf << 3);           // C/D: row = vgpr + 8*(lane>=16)
        y[(size_t)m0 * Cout + co]        = acc0[v] + bv;
        y[(size_t)(m0 + 16) * Cout + co] = acc1[v] + bv;
    }
}

// ---------------------------------------------------------------------------
// Deterministic two-stage BN statistics (sum, sum of squares per channel).
// ---------------------------------------------------------------------------
__global__ void bn_stats1(const float* __restrict__ y, float* __restrict__ part, int C)
{
    const int c    = threadIdx.x;                  // blockDim.x == C (power of 2)
    const int rows = MTOT / gridDim.x;
    const int r0   = blockIdx.x * rows;
    float s = 0.f, s2 = 0.f;
    for (int r = r0; r < r0 + rows; ++r) {
        const float v = y[(size_t)r * C + c];
        s += v; s2 += v * v;
    }
    part[(size_t)blockIdx.x * 2 * C + c]     = s;
    part[(size_t)blockIdx.x * 2 * C + C + c] = s2;
}

__global__ void bn_stats2(const float* __restrict__ part, float* __restrict__ sums,
                          int C, int nblk)
{
    const int c = threadIdx.x;
    float s = 0.f, s2 = 0.f;
    for (int i = 0; i < nblk; ++i) {
        s  += part[(size_t)i * 2 * C + c];
        s2 += part[(size_t)i * 2 * C + C + c];
    }
    sums[c]     = s;
    sums[C + c] = s2;
}

// ---------------------------------------------------------------------------
// In-place BN (training stats, biased var, eps=1e-5) + ReLU.
// ---------------------------------------------------------------------------
__global__ void bn_relu_kernel(float* __restrict__ y, const float* __restrict__ sums,
                               const float* __restrict__ g, const float* __restrict__ be,
                               int C)
{
    const float invN = 1.0f / (float)MTOT;
    const size_t total = (size_t)MTOT * C;
    for (size_t idx = (size_t)blockIdx.x * blockDim.x + threadIdx.x; idx < total;
         idx += (size_t)gridDim.x * blockDim.x) {
        const int   c    = (int)(idx & (size_t)(C - 1));
        const float mean = sums[c] * invN;
        const float var  = sums[C + c] * invN - mean * mean;
        const float s    = g[c] * rsqrtf(var + 1e-5f);
        const float sh   = be[c] - mean * s;
        const float v    = y[idx] * s + sh;
        y[idx] = fmaxf(v, 0.0f);
    }
}

// BN + residual add + ReLU -> out
__global__ void bn_add_relu_kernel(const float* __restrict__ y, const float* __restrict__ res,
                                   const float* __restrict__ sums,
                                   const float* __restrict__ g, const float* __restrict__ be,
                                   float* __restrict__ out, int C)
{
    const float invN = 1.0f / (float)MTOT;
    const size_t total = (size_t)MTOT * C;
    for (size_t idx = (size_t)blockIdx.x * blockDim.x + threadIdx.x; idx < total;
         idx += (size_t)gridDim.x * blockDim.x) {
        const int   c    = (int)(idx & (size_t)(C - 1));
        const float mean = sums[c] * invN;
        const float var  = sums[C + c] * invN - mean * mean;
        const float s    = g[c] * rsqrtf(var + 1e-5f);
        const float sh   = be[c] - mean * s;
        const float v    = y[idx] * s + sh + res[idx];
        out[idx] = fmaxf(v, 0.0f);
    }
}

// ---------------------------------------------------------------------------
// Attention head + logits + softmax + argmax. One wave32 per time row.
// logits = sigmoid(z.wa + ba) * (z @ wl) + bl
// ---------------------------------------------------------------------------
__global__ __launch_bounds__(128) void head_kernel(
    const float* __restrict__ z, const float* __restrict__ wa, const float* __restrict__ ba,
    const float* __restrict__ wl, const float* __restrict__ bl,
    float* __restrict__ logits, float* __restrict__ probs, int* __restrict__ preds)
{
    const int lane = threadIdx.x & 31;
    const int wv   = threadIdx.x >> 5;
    const int row  = (blockIdx.x << 2) + wv;
    const float* zr = z + (size_t)row * 128;

    float da = 0.f, d0 = 0.f, d1 = 0.f, d2 = 0.f, d3 = 0.f, d4 = 0.f;
#pragma unroll
    for (int k = 0; k < 4; ++k) {
        const int c = lane + 32 * k;
        const float zv = zr[c];
        da += zv * wa[c];
        const float* wr = wl + c * NCLS;
        d0 += zv * wr[0]; d1 += zv * wr[1]; d2 += zv * wr[2];
        d3 += zv * wr[3]; d4 += zv * wr[4];
    }
#pragma unroll
    for (int off = 16; off >= 1; off >>= 1) {
        da += __shfl_xor(da, off, 32);
        d0 += __shfl_xor(d0, off, 32);
        d1 += __shfl_xor(d1, off, 32);
        d2 += __shfl_xor(d2, off, 32);
        d3 += __shfl_xor(d3, off, 32);
        d4 += __shfl_xor(d4, off, 32);
    }
    if (lane == 0) {
        const float attn = 1.0f / (1.0f + expf(-(da + ba[0])));
        float lg[NCLS] = { attn * d0 + bl[0], attn * d1 + bl[1], attn * d2 + bl[2],
                           attn * d3 + bl[3], attn * d4 + bl[4] };
        float mx = lg[0];
        int   am = 0;
#pragma unroll
        for (int j = 1; j < NCLS; ++j) if (lg[j] > mx) { mx = lg[j]; am = j; }
        float ex[NCLS], ssum = 0.f;
#pragma unroll
        for (int j = 0; j < NCLS; ++j) { ex[j] = expf(lg[j] - mx); ssum += ex[j]; }
        const float inv = 1.0f / ssum;
#pragma unroll
        for (int j = 0; j < NCLS; ++j) {
            logits[(size_t)row * NCLS + j] = lg[j];
            probs[(size_t)row * NCLS + j]  = ex[j] * inv;
        }
        preds[row] = am;
    }
}

// Windowed counts over raw preds (window [t-2, t+3) within each batch row).
__global__ void counts_kernel(const int* __restrict__ preds, int* __restrict__ st,
                              int* __restrict__ sd, int* __restrict__ si)
{
    const int i = blockIdx.x * blockDim.x + threadIdx.x;
    if (i >= MTOT) return;
    const int bidx = i >> LOG2L, t = i & (LEN - 1);
    const int s = (t - 2 < 0) ? 0 : t - 2;
    const int e = (t + 3 > LEN) ? LEN : t + 3;
    int a = 0, b = 0, c = 0;
    for (int j = s; j < e; ++j) {
        const int p = preds[(bidx << LOG2L) + j];
        a += (p <= 1); b += (p == 0); c += (p == 1);
    }
    st[i] = a; sd[i] = b; si[i] = c;
}

// Pass 1: per-batch sequential smoothing (16 independent scans).
__global__ void pass1_kernel(const int* __restrict__ preds, const float* __restrict__ probs,
                             const int* __restrict__ stc, const int* __restrict__ sdc,
                             const int* __restrict__ sic, int* __restrict__ p1)
{
    const int bidx = threadIdx.x;
    if (bidx >= NBATCH) return;
    int c = 0, prev_pred = -1, prev_out = -1;
    for (int t = 0; t < LEN; ++t) {
        const int i = (bidx << LOG2L) + t;
        const int cur = preds[i];
        float pr[NCLS];
#pragma unroll
        for (int j = 0; j < NCLS; ++j) pr[j] = probs[(size_t)i * NCLS + j];

        c = (prev_pred == 3 && cur == 3) ? c + 1 : ((cur == 3) ? 1 : 0);

        const int   pc        = prev_out < 0 ? 0 : (prev_out > 4 ? 4 : prev_out);
        const float prev_prob = pr[pc];
        float q[NCLS];
#pragma unroll
        for (int j = 0; j < NCLS; ++j) q[j] = pr[j];
        q[3] = 0.0f;
        int argA = 0; float bm = q[0];
#pragma unroll
        for (int j = 1; j < NCLS; ++j) if (q[j] > bm) { bm = q[j]; argA = j; }

        int outv = cur;
        if (cur == 3 && t > 0 && (prev_out == 0 || prev_out == 1)) {
            outv = (pr[3] < 0.95f) ? ((prev_prob > 0.2f) ? prev_out : argA) : cur;
        }
        const int s = (t - 2 < 0) ? 0 : t - 2;
        const int e = (t + 3 > LEN) ? LEN : t + 3;
        const int wlen = e - s;
        const bool maj = (float)stc[i] > (float)wlen * 0.4f;
        const int smooth = (sdc[i] >= sic[i]) ? 0 : 1;
        if (cur == 3 && maj) outv = smooth;
        else if (cur == 3 && !maj && c < 3 && t > 0) outv = prev_out;

        p1[i] = outv;
        prev_pred = cur;
        prev_out  = outv;
    }
}

// Pass 2: transition rules. dur carries across the flattened tensor -> single thread.
__global__ void pass2_kernel(const int* __restrict__ p1, const float* __restrict__ probs,
                             int* __restrict__ out)
{
    const float T[NCLS][NCLS] = {
        { 1.f,  1.f,  1.f, 0.3f, 1.f },
        { 1.f,  1.f,  0.f, 0.3f, 1.f },
        { 0.f,  1.f,  1.f, 1.f,  1.f },
        { 0.7f, 0.7f, 1.f, 1.f,  1.f },
        { 1.f,  1.f,  1.f, 1.f,  1.f } };
    int prev_out = -1, prev_sv = -1, dur = 0;
    for (int i = 0; i < MTOT; ++i) {
        const int t    = i & (LEN - 1);
        const int cur0 = p1[i];
        float pr[NCLS];
#pragma unroll
        for (int j = 0; j < NCLS; ++j) pr[j] = probs[(size_t)i * NCLS + j];
        const int   prev_c = prev_out < 0 ? 0 : (prev_out > 4 ? 4 : prev_out);
        const float tp     = T[prev_c][cur0];
        const int cur_frac = (pr[cur0] < tp) ? prev_out : cur0;
        float vsum = 0.f, am = -1.f; int az = 0;
#pragma unroll
        for (int j = 0; j < NCLS; ++j) {
            const float vv = (T[prev_c][j] == 0.f) ? 0.f : pr[j];
            vsum += vv;
            if (vv > am) { am = vv; az = j; }
        }
        const int cur_zero = (vsum > 0.f) ? az : prev_out;
        const int cur_p3   = (dur < 20) ? prev_out : cur0;
        const int cur_neq  = (prev_out == 3) ? cur_p3
                           : ((tp > 0.f && tp < 1.f) ? cur_frac
                           : ((tp == 0.f) ? cur_zero : cur0));
        const int cur = (t > 0 && cur0 != prev_out) ? cur_neq : cur0;
        prev_sv = (t > 0) ? prev_out : prev_sv;
        dur = (cur == 3) ? dur + 1 : ((prev_sv == 3) ? 0 : dur);
        out[i] = cur;
        prev_out = cur;
    }
}

// ---------------------------------------------------------------------------
extern "C" void kernel_launch(void* const* d_in, const int* in_sizes, int n_in,
                              void* d_out, int out_size, void* d_ws, size_t ws_size,
                              hipStream_t stream)
{
    (void)in_sizes; (void)n_in; (void)out_size; (void)ws_size;
    const float* x = (const float*)d_in[0];

    struct Layer {
        const float *w1, *b1, *g1, *be1, *w2, *b2, *g2, *be2, *wd, *bd;
        int cout, dil;
    } L[4];
    L[0] = { (const float*)d_in[1],  (const float*)d_in[2],  (const float*)d_in[3],
             (const float*)d_in[4],  (const float*)d_in[5],  (const float*)d_in[6],
             (const float*)d_in[7],  (const float*)d_in[8],  (const float*)d_in[9],
             (const float*)d_in[10], 64, 1 };
    L[1] = { (const float*)d_in[11], (const float*)d_in[12], (const float*)d_in[13],
             (const float*)d_in[14], (const float*)d_in[15], (const float*)d_in[16],
             (const float*)d_in[17], (const float*)d_in[18], (const float*)d_in[19],
             (const float*)d_in[20], 128, 2 };
    L[2] = { (const float*)d_in[21], (const float*)d_in[22], (const float*)d_in[23],
             (const float*)d_in[24], (const float*)d_in[25], (const float*)d_in[26],
             (const float*)d_in[27], (const float*)d_in[28], nullptr, nullptr, 128, 4 };
    L[3] = { (const float*)d_in[29], (const float*)d_in[30], (const float*)d_in[31],
             (const float*)d_in[32], (const float*)d_in[33], (const float*)d_in[34],
             (const float*)d_in[35], (const float*)d_in[36], nullptr, nullptr, 128, 8 };
    const float* wa = (const float*)d_in[37];
    const float* ba = (const float*)d_in[38];
    const float* wl = (const float*)d_in[39];
    const float* bl = (const float*)d_in[40];

    // Workspace layout (floats). 4x 33.5MB activation buffers stay L2-resident (192MB L2).
    float* ws = (float*)d_ws;
    const size_t SBUF = (size_t)MTOT * 128;
    float* buf[4] = { ws, ws + SBUF, ws + 2 * SBUF, ws + 3 * SBUF };
    float* part   = ws + 4 * SBUF;            // 256 * 2 * 128
    float* sums   = part + 256 * 2 * 128;     // 2 * 128
    float* probs  = sums + 2 * 128;           // MTOT * 5
    int*   preds  = (int*)(probs + (size_t)MTOT * NCLS);
    int*   p1     = preds + MTOT;
    int*   stc    = p1 + MTOT;
    int*   sdc    = stc + MTOT;
    int*   sic    = sdc + MTOT;

    float* logits  = (float*)d_out;
    int*   outpred = (int*)((float*)d_out + (size_t)MTOT * NCLS);

    // Buffer schedule: h -> t1 -> (bn,relu) -> t2 ; R = conv1x1(h) or h ; fuse -> hout
    const float* hptr[4] = { x,      buf[2], buf[2], buf[3] };
    float*       t1      = buf[0];
    float*       t2      = buf[1];
    float*       rb      = buf[3];
    float*       hout[4] = { buf[2], buf[2], buf[3], buf[2] };

    const int GX = MTOT / 128;   // 512: 32 rows/wave * 4 waves/block

    for (int i = 0; i < 4; ++i) {
        const Layer& P = L[i];
        const dim3 g(GX, P.cout / 16);
        // conv1 + optional 1x1 residual conv (compile-time CIN dispatch)
        const float* R = hptr[i];
        if (i == 0) {
            conv_wmma_kernel<3, 34><<<g, 128, 0, stream>>>(hptr[i], P.w1, P.b1, t1, P.cout, P.dil);
            conv_wmma_kernel<1, 34><<<g, 128, 0, stream>>>(hptr[i], P.wd, P.bd, rb, P.cout, 1);
            R = rb;
        } else if (i == 1) {
            conv_wmma_kernel<3, 64><<<g, 128, 0, stream>>>(hptr[i], P.w1, P.b1, t1, P.cout, P.dil);
            conv_wmma_kernel<1, 64><<<g, 128, 0, stream>>>(hptr[i], P.wd, P.bd, rb, P.cout, 1);
            R = rb;
        } else {
            conv_wmma_kernel<3, 128><<<g, 128, 0, stream>>>(hptr[i], P.w1, P.b1, t1, P.cout, P.dil);
        }
        // BN1 stats + BN+ReLU (in place)
        bn_stats1<<<256, P.cout, 0, stream>>>(t1, part, P.cout);
        bn_stats2<<<1, P.cout, 0, stream>>>(part, sums, P.cout, 256);
        bn_relu_kernel<<<2048, 256, 0, stream>>>(t1, sums, P.g1, P.be1, P.cout);
        // conv2 (CIN == cout of this layer)
        if (i == 0) {
            conv_wmma_kernel<3, 64><<<g, 128, 0, stream>>>(t1, P.w2, P.b2, t2, P.cout, P.dil);
        } else {
            conv_wmma_kernel<3, 128><<<g, 128, 0, stream>>>(t1, P.w2, P.b2, t2, P.cout, P.dil);
        }
        // BN2 stats + BN + residual + ReLU -> hout
        bn_stats1<<<256, P.cout, 0, stream>>>(t2, part, P.cout);
        bn_stats2<<<1, P.cout, 0, stream>>>(part, sums, P.cout, 256);
        bn_add_relu_kernel<<<2048, 256, 0, stream>>>(t2, R, sums, P.g2, P.be2,
                                                     hout[i], P.cout);
    }

    const float* z = hout[3];   // [MTOT, 128]
    head_kernel<<<MTOT / 4, 128, 0, stream>>>(z, wa, ba, wl, bl, logits, probs, preds);
    counts_kernel<<<MTOT / 256, 256, 0, stream>>>(preds, stc, sdc, sic);
    pass1_kernel<<<1, 32, 0, stream>>>(preds, probs, stc, sdc, sic, p1);
    pass2_kernel<<<1, 1, 0, stream>>>(p1, probs, outpred);
}